// QCQP_cvxpy_25529285607576
// MI455X (gfx1250) — compile-verified
//
#include <hip/hip_runtime.h>
#include <math.h>

// Problem constants (from reference)
#define NCONE        8
#define NN           24
#define REGEPS       1e-7f
#define BATCH        32768
#define FISTA_ITERS  200
#define POWER_ITERS  30

#define PROB_PER_BLOCK 32
#define G_STRIDE 580   // dwords/problem: 576 + 4 pad -> 16B-aligned rows, good b128 LDS access
#define B_STRIDE 25    // dwords/problem for b vector (odd stride -> conflict-free)
#define ZSLOT    608   // index into sA of a guaranteed-zero dword (broadcast read)

static_assert(BATCH % PROB_PER_BLOCK == 0, "exact grid");

typedef float v2f __attribute__((ext_vector_type(2)));
typedef float v8f __attribute__((ext_vector_type(8)));

// 24x24 matvec, G resident in LDS, read as float4 -> ds_load_b128 (16B-aligned by construction)
__device__ __forceinline__ void matvec24(const float4* __restrict__ G4,
                                         const float* __restrict__ v,
                                         float* __restrict__ w) {
#pragma unroll
  for (int i = 0; i < NN; ++i) {
    float acc = 0.f;
#pragma unroll
    for (int j = 0; j < NN / 4; ++j) {
      float4 g = G4[i * (NN / 4) + j];
      acc = fmaf(g.x, v[4 * j + 0], acc);
      acc = fmaf(g.y, v[4 * j + 1], acc);
      acc = fmaf(g.z, v[4 * j + 2], acc);
      acc = fmaf(g.w, v[4 * j + 3], acc);
    }
    w[i] = acc;
  }
}

__launch_bounds__(PROB_PER_BLOCK, 1)
__global__ void qcqp_fused_kernel(const float* __restrict__ P,
                                  const float* __restrict__ q,
                                  float* __restrict__ out) {
  __shared__ __align__(16) float sG[PROB_PER_BLOCK * G_STRIDE];   // 74,240 B  (Gram matrices)
  __shared__ __align__(16) float sB[PROB_PER_BLOCK * B_STRIDE];   //  3,200 B  (b vectors, later reused for output transpose)
  __shared__ __align__(16) float sA[640];                         //  2,560 B  staging: A(576) + q(24) + zero pad
  __shared__ float sDump[32];                                     //    128 B  sink for masked-off stores

  const int lane = threadIdx.x;   // 0..31 (wave32)
  const int blk  = blockIdx.x;
  const int ll   = lane & 15;     // lane within half-wave
  const int lh   = lane >> 4;     // half-wave id

  // zero slot for out-of-range operand reads (branchless padding)
  sA[ZSLOT + lane] = 0.f;

  // Per-lane constant operand/scatter addressing decisions (selects, not branches)
  const int  n1      = ll + 16;          // B-op tile1 global column 16..31 (24 == q column)
  const bool m1ok    = (ll + 16) < NN;   // A-op tile1 row valid
  float* const dmp   = &sDump[lane];

  // ---------------- Phase 1: G = (P + eps I)^T (P + eps I), b = -(A^T q - eps q), via WMMA f32 16x16x4 ----------------
#pragma unroll 1
  for (int p = 0; p < PROB_PER_BLOCK; ++p) {
    const long pg = (long)blk * PROB_PER_BLOCK + p;
    const float4* Pp4 = (const float4*)(P + pg * (NN * NN));   // 2304 B/problem -> 16B aligned
    const float*  qp  = q + pg * NN;

    // Stage A = P into LDS with b128 loads/stores (144 float4), then q, then pre-add diagonal eps.
    float4* sA4 = (float4*)sA;
#pragma unroll
    for (int t = 0; t < 4; ++t) sA4[t * 32 + lane] = Pp4[t * 32 + lane];
    if (lane < 16) sA4[128 + lane] = Pp4[128 + lane];
    if (lane < NN) {
      sA[NN * NN + lane] = qp[lane];       // q at [576..599]
      sA[lane * 25] += REGEPS;             // diag element d*24+d = 25d  (LDS in-order within the wave)
    }

    // D = C x Bmat ; C = A^T (rows 24..31 zero-padded), Bmat = [A | q | 0]
    v8f acc00 = {}; v8f acc01 = {}; v8f acc10 = {}; v8f acc11 = {};

#pragma unroll
    for (int kc = 0; kc < 6; ++kc) {
      const int k    = kc * 4 + 2 * lh;    // this half-wave supplies K = k, k+1
      const int base = k * NN;

      // tile0 operand (rows/cols 0..15): A-op value A[k][ll] == B-op value -> shared loads
      const float c0x = sA[base + ll];
      const float c0y = sA[base + NN + ll];
      // A-op tile1 (rows 16..31): address-select into zero slot when row >= 24
      const int a1ix = m1ok ? (base + 16 + ll)      : ZSLOT;
      const int a1iy = m1ok ? (base + NN + 16 + ll) : ZSLOT;
      const float a1x = sA[a1ix];
      const float a1y = sA[a1iy];
      // B-op tile1 (cols 16..31): col<24 -> A, col 24 -> q, else zero
      const int b1ix = (n1 < NN) ? (base + n1)      : ((n1 == NN) ? (NN * NN + k)     : ZSLOT);
      const int b1iy = (n1 < NN) ? (base + NN + n1) : ((n1 == NN) ? (NN * NN + k + 1) : ZSLOT);
      const float b1x = sA[b1ix];
      const float b1y = sA[b1iy];

      const v2f A0 = {c0x, c0y}, A1 = {a1x, a1y};
      const v2f B0 = {c0x, c0y}, B1 = {b1x, b1y};

      acc00 = __builtin_amdgcn_wmma_f32_16x16x4_f32(false, A0, false, B0, (short)0, acc00, false, false);
      acc01 = __builtin_amdgcn_wmma_f32_16x16x4_f32(false, A0, false, B1, (short)0, acc01, false, false);
      acc10 = __builtin_amdgcn_wmma_f32_16x16x4_f32(false, A1, false, B0, (short)0, acc10, false, false);
      acc11 = __builtin_amdgcn_wmma_f32_16x16x4_f32(false, A1, false, B1, (short)0, acc11, false, false);
    }

    // Scatter D tiles -> sG / sB, branchless via address selects into the dump slot.
    // C/D layout: lanes 0-15: M = r ; lanes 16-31: M = r + 8 ; N = lane&15.
    float* const Gp = &sG[p * G_STRIDE];
    float* const bp = &sB[p * B_STRIDE];
#pragma unroll
    for (int r = 0; r < 8; ++r) {
      const int m  = r + 8 * lh;           // 0..15
      const int m1 = 16 + r;               // valid row for lh==0 only

      Gp[m * NN + ll] = acc00[r];                                          // tile (0,0)

      float* t01 = (ll < 8) ? &Gp[m * NN + 16 + ll] : dmp;                 // tile (0,1) cols 16..23
      *t01 = acc01[r];
      const float qm = sA[NN * NN + m];
      float* pb = (ll == 8) ? &bp[m] : dmp;                                // col 24 -> b
      *pb = -(acc01[r] - REGEPS * qm);

      float* t10 = (lh == 0) ? &Gp[m1 * NN + ll] : dmp;                    // tile (1,0) rows 16..23
      *t10 = acc10[r];
      float* t11 = (lh == 0 && ll < 8) ? &Gp[m1 * NN + 16 + ll] : dmp;     // tile (1,1)
      *t11 = acc11[r];
      const float qm1 = sA[NN * NN + 16 + r];
      float* pb1 = (lh == 0 && ll == 8) ? &bp[16 + r] : dmp;
      *pb1 = -(acc11[r] - REGEPS * qm1);
    }
  }
  __syncthreads();  // single-wave WG -> no-op, kept for clarity

  // ---------------- Phase 2: per-lane power iteration + FISTA (one problem per lane) ----------------
  const float4* G4 = (const float4*)&sG[lane * G_STRIDE];   // 2320 B stride -> 16B aligned

  float bb[NN];
#pragma unroll
  for (int i = 0; i < NN; ++i) bb[i] = sB[lane * B_STRIDE + i];

  // power iteration for Lipschitz constant
  float v[NN], w[NN];
#pragma unroll
  for (int i = 0; i < NN; ++i) v[i] = 0.20412414523193154f;   // 1/sqrt(24)
#pragma unroll 1
  for (int it = 0; it < POWER_ITERS; ++it) {
    matvec24(G4, v, w);
    float s = 0.f;
#pragma unroll
    for (int i = 0; i < NN; ++i) s = fmaf(w[i], w[i], s);
    const float inv = 1.0f / (sqrtf(s) + 1e-30f);
#pragma unroll
    for (int i = 0; i < NN; ++i) v[i] = w[i] * inv;
  }
  matvec24(G4, v, w);
  float Lv = 0.f;
#pragma unroll
  for (int i = 0; i < NN; ++i) Lv = fmaf(v[i], w[i], Lv);
  const float step = 1.0f / (Lv + 1e-12f);

  // FISTA
  float l[NN], y[NN], ln[NN];
#pragma unroll
  for (int i = 0; i < NN; ++i) { l[i] = 0.f; y[i] = 0.f; }
  float tk = 1.0f;

#pragma unroll 1
  for (int it = 0; it < FISTA_ITERS; ++it) {
    matvec24(G4, y, w);                   // w = G y
    float z[NN];
#pragma unroll
    for (int i = 0; i < NN; ++i) z[i] = y[i] - step * (w[i] + bb[i]);

    // project onto product of 8 second-order cones (pure per-lane selects)
#pragma unroll
    for (int c = 0; c < NCONE; ++c) {
      const float t  = z[c];
      const float x0 = z[NCONE + 2 * c];
      const float x1 = z[NCONE + 2 * c + 1];
      const float nx = sqrtf(fmaf(x0, x0, x1 * x1));
      const float alpha  = 0.5f * (t + nx);
      const bool inside = (nx <= t);
      const bool zero   = (nx <= -t);
      const float tout  = inside ? t   : (zero ? 0.f : alpha);
      const float scale = inside ? 1.f : (zero ? 0.f : alpha / fmaxf(nx, 1e-12f));
      ln[c]                 = tout;
      ln[NCONE + 2 * c]     = x0 * scale;
      ln[NCONE + 2 * c + 1] = x1 * scale;
    }

    const float tk_new = 0.5f * (1.0f + sqrtf(fmaf(4.0f * tk, tk, 1.0f)));
    const float coef = (tk - 1.0f) / tk_new;
#pragma unroll
    for (int i = 0; i < NN; ++i) {
      y[i] = ln[i] + coef * (ln[i] - l[i]);
      l[i] = ln[i];
    }
    tk = tk_new;
  }

  // Coalesced output: transpose through LDS (reuse sB: 32*25 >= needed), then b128 stores.
#pragma unroll
  for (int i = 0; i < NN; ++i) sB[lane * B_STRIDE + i] = l[i];
  float4* out4 = (float4*)(out + (long)blk * (PROB_PER_BLOCK * NN));
#pragma unroll
  for (int t = 0; t < (PROB_PER_BLOCK * NN) / (4 * 32); ++t) {   // 6 iterations
    const int f4 = t * 32 + lane;
    const int f  = 4 * f4;
    float4 o;
    o.x = sB[(f + 0) / NN * B_STRIDE + (f + 0) % NN];
    o.y = sB[(f + 1) / NN * B_STRIDE + (f + 1) % NN];
    o.z = sB[(f + 2) / NN * B_STRIDE + (f + 2) % NN];
    o.w = sB[(f + 3) / NN * B_STRIDE + (f + 3) % NN];
    out4[f4] = o;
  }
}

extern "C" void kernel_launch(void* const* d_in, const int* in_sizes, int n_in,
                              void* d_out, int out_size, void* d_ws, size_t ws_size,
                              hipStream_t stream) {
  (void)in_sizes; (void)n_in; (void)d_ws; (void)ws_size; (void)out_size;
  const float* P = (const float*)d_in[0];   // [B, 24, 24] f32
  const float* q = (const float*)d_in[1];   // [B, 24, 1]  f32
  float* out = (float*)d_out;               // [B, 24]     f32
  qcqp_fused_kernel<<<BATCH / PROB_PER_BLOCK, PROB_PER_BLOCK, 0, stream>>>(P, q, out);
}